// RipsH1_62775241999084
// MI455X (gfx1250) — compile-verified
//
#include <hip/hip_runtime.h>

typedef float v2f __attribute__((ext_vector_type(2)));
typedef float v8f __attribute__((ext_vector_type(8)));

// Each wave handles 16 output pairs.
// For every K-chunk of 4, lane l holds Diff[m][koff..koff+1] with
//   m = l & 15, koff = (l>>4)*2   -> exactly the documented f32 16x16x4 A layout.
// Passing the same registers as A and B gives D = Diff * Diff^T (Gram);
// diag(D) = squared distances.  Diagonal (i,i) lives at:
//   i < 8 : lane i,      VGPR i
//   i >= 8: lane i + 16, VGPR i - 8
__global__ __launch_bounds__(128) void rips_dist_wmma_kernel(
    const float* __restrict__ input,
    const long long* __restrict__ verts,   // int64 indices, flat (N_GEN*4)
    float* __restrict__ out,               // nPairs floats
    int nPairs)
{
    const int lane        = threadIdx.x & 31;
    const int waveInBlock = threadIdx.x >> 5;
    const long long waveId = (long long)blockIdx.x * (blockDim.x >> 5) + waveInBlock;
    const long long Pbase  = waveId * 16;

    const int m    = lane & 15;          // pair-in-wave this lane feeds
    const int koff = (lane >> 4) << 1;   // 0 for lanes 0-15, 2 for lanes 16-31

    // Pair P uses verts[2P] (a-index) and verts[2P+1] (b-index):
    //   P = 2g+j  ->  2P = 4g+2j = verts[g][2j] = {v0,v2},  2P+1 = {v1,v3}
    long long P = Pbase + m;
    if (P >= nPairs) P = nPairs - 1;     // clamp (no partial waves at 1M, but safe)

    const long long ia = verts[2 * P];
    const long long ib = verts[2 * P + 1];

    const float* pa = input + ia * 64 + koff;
    const float* pb = input + ib * 64 + koff;

    v8f c0 = {};
    v8f c1 = {};
#pragma unroll
    for (int kc = 0; kc < 16; kc += 2) {
        v2f a0 = *(const v2f*)(pa + kc * 4);
        v2f b0 = *(const v2f*)(pb + kc * 4);
        v2f a1 = *(const v2f*)(pa + kc * 4 + 4);
        v2f b1 = *(const v2f*)(pb + kc * 4 + 4);
        v2f d0 = a0 - b0;
        v2f d1 = a1 - b1;
        // 8 args: (neg_a, A, neg_b, B, c_mod, C, reuse_a, reuse_b)
        c0 = __builtin_amdgcn_wmma_f32_16x16x4_f32(false, d0, false, d0,
                                                   (short)0, c0, false, false);
        c1 = __builtin_amdgcn_wmma_f32_16x16x4_f32(false, d1, false, d1,
                                                   (short)0, c1, false, false);
    }

    // Extract diagonal: only lanes 0-7 and 24-31 hold diagonal elements.
    const bool isDiag = (lane < 8) || (lane >= 24);
    const int  r      = lane & 7;        // which accumulator VGPR holds diag

    float s0 = (r == 0) ? c0[0] : (r == 1) ? c0[1] : (r == 2) ? c0[2]
             : (r == 3) ? c0[3] : (r == 4) ? c0[4] : (r == 5) ? c0[5]
             : (r == 6) ? c0[6] : c0[7];
    float s1 = (r == 0) ? c1[0] : (r == 1) ? c1[1] : (r == 2) ? c1[2]
             : (r == 3) ? c1[3] : (r == 4) ? c1[4] : (r == 5) ? c1[5]
             : (r == 6) ? c1[6] : c1[7];
    const float sq = s0 + s1;

    if (isDiag) {
        const int mo = (lane < 8) ? lane : (lane - 16);   // pair index in wave
        const long long Pout = Pbase + mo;
        if (Pout < nPairs) out[Pout] = __builtin_sqrtf(sq);
    }
}

extern "C" void kernel_launch(void* const* d_in, const int* in_sizes, int n_in,
                              void* d_out, int out_size, void* d_ws, size_t ws_size,
                              hipStream_t stream) {
    (void)in_sizes; (void)n_in; (void)d_ws; (void)ws_size;

    const float*     input = (const float*)d_in[0];       // (100000, 64) f32
    const long long* verts = (const long long*)d_in[1];   // (500000, 4) i64
    float*           out   = (float*)d_out;               // 1,000,000 f32

    const int nPairs        = out_size;                   // 1,000,000
    const int pairsPerWave  = 16;
    const int block         = 128;                        // 4 wave32 per block
    const int wavesTotal    = (nPairs + pairsPerWave - 1) / pairsPerWave;
    const int wavesPerBlock = block / 32;
    const int grid          = (wavesTotal + wavesPerBlock - 1) / wavesPerBlock;

    rips_dist_wmma_kernel<<<grid, block, 0, stream>>>(input, verts, out, nPairs);
}